// RegularizationModule_33397665694036
// MI455X (gfx1250) — compile-verified
//
#include <hip/hip_runtime.h>

// ---------------------------------------------------------------------------
// Segment-softmax message passing (graph attention-like) for MI455X (gfx1250).
//
// out[d] = sum_{e->d} (R_src @ pos_src + trans_src) * softmax_over_dst(w_e)
//   w_e = relu(x[src][15] - x[dst][15] - 0.1), edges = E real + N self-loops.
//
// Memory-bound (23.3 TB/s HBM, 192 MB L2): x (12.8 MB) + pos (2.4 MB) are
// L2-resident gather tables; the 51 MB edge stream is read twice.
//   Pass 1: per-dst max of w via u32-bit atomicMax (w >= 0; every segment's
//           self-loop contributes exactly 0 == init, so only real edges run).
//   Pass 2: real edges only; e = exp(w - m[dst]); global_atomic_add_f32 of
//           {v*e, e} into a float4 accumulator per node (L2-resolved).
//   Pass 3: fold the per-node self-loop message in (atomic-free, coalesced)
//           and normalize: out = (acc.xyz + v_self*e_self)/(acc.w+e_self+eps).
//
// Both edge passes stage the src/dst index stream through LDS with per-wave
// double-buffered global_load_async_to_lds_b32 + s_wait_asynccnt: deep
// software prefetch at zero VGPR-destination cost, wave-local (lane i reads
// only its own LDS slot -> no barriers), overlapped with the random gathers
// and L2 atomics that dominate.
//
// WMMA intentionally NOT used: per-edge *distinct* 3x3 matrices cannot batch
// into a shared-B 16x16 WMMA (<=9/1024 useful MACs) -- VALU FMAs are free here.
// ---------------------------------------------------------------------------

__global__ void rm_init_kernel(float4* __restrict__ acc, float* __restrict__ m, int n) {
    int i = blockIdx.x * blockDim.x + threadIdx.x;
    if (i < n) {
        acc[i] = make_float4(0.f, 0.f, 0.f, 0.f);
        m[i]   = 0.f;   // == segment max lower bound (self-loop w = 0)
    }
}

// Issue one 32-edge tile's src+dst indices into LDS buffer (ASYNCcnt += 2).
// GV-mode per-lane addressing; clamped (safe) prefetch at the tail.
__device__ __forceinline__ void rm_issue_tile(const int* __restrict__ src,
                                              const int* __restrict__ dst,
                                              int* base, int buf,
                                              int tile, int lane, int e) {
    long idx = (long)tile * 32 + lane;
    if (idx > (long)e - 1) idx = (long)e - 1;
    unsigned lsrc = (unsigned)(size_t)&base[buf * 64 + lane];
    unsigned ldst = (unsigned)(size_t)&base[buf * 64 + 32 + lane];
    asm volatile("global_load_async_to_lds_b32 %0, %1, off"
                 :: "v"(lsrc), "v"(&src[idx]) : "memory");
    asm volatile("global_load_async_to_lds_b32 %0, %1, off"
                 :: "v"(ldst), "v"(&dst[idx]) : "memory");
}

// Pass 1: segment max of w over dst (real edges only), async-staged stream.
__global__ void __launch_bounds__(256)
rm_max_kernel(const float* __restrict__ x,
              const int* __restrict__ src,
              const int* __restrict__ dst,
              unsigned int* __restrict__ m_bits, int e) {
    __shared__ int sh[8 * 128];           // 8 waves * 2 bufs * 64 ints = 4 KB
    const int lane   = threadIdx.x & 31;
    const int wlocal = threadIdx.x >> 5;
    const int wglob  = (int)((blockIdx.x * blockDim.x + threadIdx.x) >> 5);
    const int nw     = (int)((gridDim.x * blockDim.x) >> 5);
    const int ntiles = (e + 31) >> 5;
    int* base = &sh[wlocal * 128];

    if (wglob >= ntiles) return;          // wave-uniform exit

    int buf = 0;
    rm_issue_tile(src, dst, base, 0, wglob, lane, e);      // ASYNCcnt = 2
    for (int t = wglob; t < ntiles; t += nw) {
        int tn = t + nw;
        if (tn >= ntiles) tn = ntiles - 1;
        rm_issue_tile(src, dst, base, buf ^ 1, tn, lane, e);  // 2 -> 4
        asm volatile("s_wait_asynccnt 0x2" ::: "memory");     // current landed

        int s = base[buf * 64 + lane];
        int d = base[buf * 64 + 32 + lane];
        long idx = (long)t * 32 + lane;

        float w = x[(size_t)s * 16 + 15] - x[(size_t)d * 16 + 15] - 0.1f;
        if (idx < (long)e && w > 0.f) {
            // Non-negative IEEE floats are order-isomorphic to their u32 bits.
            atomicMax(&m_bits[d], __float_as_uint(w));
        }
        buf ^= 1;
    }
}

// Pass 2: real edges; gather + exp + scatter-add, async-staged index stream.
__global__ void __launch_bounds__(256)
rm_acc_kernel(const float* __restrict__ x,
              const float* __restrict__ pos,
              const int* __restrict__ src,
              const int* __restrict__ dst,
              const float* __restrict__ m,
              float* __restrict__ acc,  // float4 per node
              int e) {
    __shared__ int sh[8 * 128];
    const int lane   = threadIdx.x & 31;
    const int wlocal = threadIdx.x >> 5;
    const int wglob  = (int)((blockIdx.x * blockDim.x + threadIdx.x) >> 5);
    const int nw     = (int)((gridDim.x * blockDim.x) >> 5);
    const int ntiles = (e + 31) >> 5;
    int* base = &sh[wlocal * 128];

    if (wglob >= ntiles) return;

    int buf = 0;
    rm_issue_tile(src, dst, base, 0, wglob, lane, e);
    for (int t = wglob; t < ntiles; t += nw) {
        int tn = t + nw;
        if (tn >= ntiles) tn = ntiles - 1;
        rm_issue_tile(src, dst, base, buf ^ 1, tn, lane, e);
        asm volatile("s_wait_asynccnt 0x2" ::: "memory");

        int s = base[buf * 64 + lane];         // ds_load (wave-local slots)
        int d = base[buf * 64 + 32 + lane];
        long idx = (long)t * 32 + lane;
        bool valid = idx < (long)e;

        // x[s] row = 64B -> 4x global_load_b128 (L2-resident gather).
        const float4* x4 = (const float4*)x;
        float4 r0 = x4[(size_t)s * 4 + 0];
        float4 r1 = x4[(size_t)s * 4 + 1];
        float4 r2 = x4[(size_t)s * 4 + 2];
        float4 r3 = x4[(size_t)s * 4 + 3];   // trans.xyz, score

        float xi15 = x[(size_t)d * 16 + 15];
        float w    = fmaxf(r3.w - xi15 - 0.1f, 0.f);
        float ew   = __expf(w - m[d]);       // w - m <= 0, safe

        float p0 = pos[(size_t)s * 3 + 0];
        float p1 = pos[(size_t)s * 3 + 1];
        float p2 = pos[(size_t)s * 3 + 2];

        // R = x[3:12] row-major 3x3.
        float v0 = fmaf(r0.w, p0, fmaf(r1.x, p1, fmaf(r1.y, p2, r3.x)));
        float v1 = fmaf(r1.z, p0, fmaf(r1.w, p1, fmaf(r2.x, p2, r3.y)));
        float v2 = fmaf(r2.y, p0, fmaf(r2.z, p1, fmaf(r2.w, p2, r3.z)));

        if (valid) {
            float* a = acc + (size_t)d * 4;
            unsafeAtomicAdd(a + 0, v0 * ew);  // global_atomic_add_f32 in L2
            unsafeAtomicAdd(a + 1, v1 * ew);
            unsafeAtomicAdd(a + 2, v2 * ew);
            unsafeAtomicAdd(a + 3, ew);
        }
        buf ^= 1;
    }
    // Trailing clamped prefetch (ASYNCcnt = 2) drains at S_ENDPGM's wait-idle.
}

// Pass 3: fold self-loop message (atomic-free, fully coalesced) + normalize.
__global__ void rm_norm_kernel(const float4* __restrict__ acc,
                               const float* __restrict__ m,
                               const float* __restrict__ x,
                               const float* __restrict__ pos,
                               float* __restrict__ out, int n) {
    int i = blockIdx.x * blockDim.x + threadIdx.x;
    if (i >= n) return;

    const float4* x4 = (const float4*)x;
    float4 r0 = x4[(size_t)i * 4 + 0];
    float4 r1 = x4[(size_t)i * 4 + 1];
    float4 r2 = x4[(size_t)i * 4 + 2];
    float4 r3 = x4[(size_t)i * 4 + 3];
    float p0 = pos[(size_t)i * 3 + 0];
    float p1 = pos[(size_t)i * 3 + 1];
    float p2 = pos[(size_t)i * 3 + 2];

    float ew = __expf(0.f - m[i]);   // self-loop: w = relu(-0.1) = 0
    float v0 = fmaf(r0.w, p0, fmaf(r1.x, p1, fmaf(r1.y, p2, r3.x)));
    float v1 = fmaf(r1.z, p0, fmaf(r1.w, p1, fmaf(r2.x, p2, r3.y)));
    float v2 = fmaf(r2.y, p0, fmaf(r2.z, p1, fmaf(r2.w, p2, r3.z)));

    float4 a = acc[i];
    float inv = 1.0f / (a.w + ew + 1e-16f);
    out[(size_t)i * 3 + 0] = fmaf(v0, ew, a.x) * inv;
    out[(size_t)i * 3 + 1] = fmaf(v1, ew, a.y) * inv;
    out[(size_t)i * 3 + 2] = fmaf(v2, ew, a.z) * inv;
}

extern "C" void kernel_launch(void* const* d_in, const int* in_sizes, int n_in,
                              void* d_out, int out_size, void* d_ws, size_t ws_size,
                              hipStream_t stream) {
    const float* x   = (const float*)d_in[0];   // (N,16) f32
    const float* pos = (const float*)d_in[1];   // (N,3)  f32
    const int*   ei  = (const int*)d_in[2];     // (2,E)  i32, row0=src row1=dst

    const int n = in_sizes[0] / 16;
    const int e = in_sizes[2] / 2;
    const int* src = ei;
    const int* dst = ei + e;

    // Workspace: acc float4[n] then m float[n]  (20*n bytes = 4 MB).
    float* acc = (float*)d_ws;
    float* m   = acc + (size_t)4 * n;
    float* out = (float*)d_out;

    const int B = 256;  // 8 wave32s per block
    // Persistent-wave tiling for the edge passes: ~24.6K waves, 200K tiles.
    int ntiles = (e + 31) >> 5;
    int edgeBlocks = (ntiles * 32 + B - 1) / B;
    if (edgeBlocks > 3072) edgeBlocks = 3072;

    rm_init_kernel<<<(n + B - 1) / B, B, 0, stream>>>((float4*)acc, m, n);
    rm_max_kernel<<<edgeBlocks, B, 0, stream>>>(x, src, dst,
                                                (unsigned int*)m, e);
    rm_acc_kernel<<<edgeBlocks, B, 0, stream>>>(x, pos, src, dst, m, acc, e);
    rm_norm_kernel<<<(n + B - 1) / B, B, 0, stream>>>((const float4*)acc, m,
                                                      x, pos, out, n);
}